// BSplineField3d_18957985644549
// MI455X (gfx1250) — compile-verified
//
#include <hip/hip_runtime.h>

// Tricubic B-spline field sampling on a 64^3 fp32 grid (MI455X / gfx1250).
// Cache-gather bound: the 1 MB table is L2-resident (192 MB L2); z-taps are
// contiguous -> one global_load_b128 gather per (l,m) row (16 loads/point).
// Reduction is fully separable (z -> m -> l) to minimize VALU per point.

#define GN   64
#define GN2  (GN * GN)
#define GN3  (GN * GN * GN)

typedef float vf4  __attribute__((ext_vector_type(4)));           // 16B aligned
typedef vf4  vf4u  __attribute__((aligned(4)));                    // 4B aligned view

__device__ __forceinline__ void bw(float f, float w[4]) {
    const float k = 1.0f / 6.0f;
    float f2 = f * f;
    float f3 = f2 * f;
    w[0] = k * (1.0f - 3.0f * f + 3.0f * f2 - f3);   // (1-f)^3/6
    w[1] = k * (3.0f * f3 - 6.0f * f2 + 4.0f);
    w[2] = k * (-3.0f * f3 + 3.0f * f2 + 3.0f * f + 1.0f);
    w[3] = k * f3;
}

__device__ __forceinline__ int clampi(int a, int hi) {
    return a < 0 ? 0 : (a > hi ? hi : a);
}

__device__ __forceinline__ float eval_point(const float* __restrict__ phi,
                                            float xv, float yv, float zv) {
    // domain [-1,1], dx = 2/(GN-3)  =>  u = (x+1) * (GN-3)/2
    const float inv_d = (float)(GN - 3) * 0.5f;   // 30.5
    float u = (xv + 1.0f) * inv_d;
    float v = (yv + 1.0f) * inv_d;
    float w = (zv + 1.0f) * inv_d;
    float fxu = floorf(u), fyv = floorf(v), fzw = floorf(w);
    int ix = (int)fxu, iy = (int)fyv, iz = (int)fzw;
    float fu = u - fxu, fv = v - fyv, fw = w - fzw;

    float wx[4], wy[4], wz[4];
    bw(fu, wx);
    bw(fv, wy);
    bw(fw, wz);

    int xo[4], yo[4];
#pragma unroll
    for (int l = 0; l < 4; ++l) xo[l] = clampi(ix + l, GN - 1) * GN2;
#pragma unroll
    for (int m = 0; m < 4; ++m) yo[m] = clampi(iy + m, GN - 1) * GN;

    float T = 0.0f;
    if (iz >= 0 && iz + 3 <= GN - 1) {
        // Fast path (always taken for x,y,z in [0,1)): z-taps contiguous,
        // one 16-byte gather per (l,m) row; separable z->m->l contraction.
#pragma unroll
        for (int l = 0; l < 4; ++l) {
            float Bl = 0.0f;
#pragma unroll
            for (int m = 0; m < 4; ++m) {
                vf4 p = *reinterpret_cast<const vf4u*>(phi + xo[l] + yo[m] + iz);
                float s = fmaf(wz[3], p.w,
                          fmaf(wz[2], p.z,
                          fmaf(wz[1], p.y, wz[0] * p.x)));
                Bl = fmaf(wy[m], s, Bl);
            }
            T = fmaf(wx[l], Bl, T);
        }
    } else {
        // General (clamped) fallback — scalar gathers; cold for given inputs.
        int zo[4];
#pragma unroll
        for (int nn = 0; nn < 4; ++nn) zo[nn] = clampi(iz + nn, GN - 1);
#pragma unroll
        for (int l = 0; l < 4; ++l) {
            float Bl = 0.0f;
#pragma unroll
            for (int m = 0; m < 4; ++m) {
                const float* b = phi + xo[l] + yo[m];
                float s = fmaf(wz[3], b[zo[3]],
                          fmaf(wz[2], b[zo[2]],
                          fmaf(wz[1], b[zo[1]], wz[0] * b[zo[0]])));
                Bl = fmaf(wy[m], s, Bl);
            }
            T = fmaf(wx[l], Bl, T);
        }
    }
    return T;
}

__global__ __launch_bounds__(256)
void bspline3d_kernel(const float* __restrict__ x,
                      const float* __restrict__ y,
                      const float* __restrict__ z,
                      const float* __restrict__ phi_x,
                      const int*   __restrict__ chan,
                      float*       __restrict__ out,
                      int n) {
    const float* __restrict__ phi = phi_x + (long)chan[0] * (long)GN3;
    int gtid    = blockIdx.x * blockDim.x + threadIdx.x;
    int gstride = gridDim.x * blockDim.x;
    int n4 = n >> 2;

    for (int g = gtid; g < n4; g += gstride) {
        // Prefetch next iteration's coordinate tiles (gfx1250 global_prefetch_b8).
        int gn = g + gstride;
        if (gn < n4) {
            __builtin_prefetch(x + 4 * gn, 0, 1);
            __builtin_prefetch(y + 4 * gn, 0, 1);
            __builtin_prefetch(z + 4 * gn, 0, 1);
        }
        // Non-temporal streaming loads: don't evict the 1 MB table from WGP$.
        vf4 xv = __builtin_nontemporal_load(reinterpret_cast<const vf4*>(x + 4 * g));
        vf4 yv = __builtin_nontemporal_load(reinterpret_cast<const vf4*>(y + 4 * g));
        vf4 zv = __builtin_nontemporal_load(reinterpret_cast<const vf4*>(z + 4 * g));

        vf4 t;
        t.x = eval_point(phi, xv.x, yv.x, zv.x);
        t.y = eval_point(phi, xv.y, yv.y, zv.y);
        t.z = eval_point(phi, xv.z, yv.z, zv.z);
        t.w = eval_point(phi, xv.w, yv.w, zv.w);

        __builtin_nontemporal_store(t, reinterpret_cast<vf4*>(out + 4 * g));
    }

    // Tail (n % 4) — handled by the first few threads.
    int tail = n & 3;
    if (gtid < tail) {
        int p = (n4 << 2) + gtid;
        out[p] = eval_point(phi, x[p], y[p], z[p]);
    }
}

extern "C" void kernel_launch(void* const* d_in, const int* in_sizes, int n_in,
                              void* d_out, int out_size, void* d_ws, size_t ws_size,
                              hipStream_t stream) {
    const float* x     = (const float*)d_in[0];
    const float* y     = (const float*)d_in[1];
    const float* z     = (const float*)d_in[2];
    const float* phi_x = (const float*)d_in[3];
    const int*   chan  = (const int*)d_in[4];
    float*       out   = (float*)d_out;
    int n = in_sizes[0];

    int groups = n >> 2;
    int blocks = (groups + 255) / 256;
    if (blocks > 2048) blocks = 2048;   // ~4 grid-stride iterations/thread at N=2^23
    if (blocks < 1) blocks = 1;

    bspline3d_kernel<<<blocks, 256, 0, stream>>>(x, y, z, phi_x, chan, out, n);
}